// HoloLinear_17222818857169
// MI455X (gfx1250) — compile-verified
//
#include <hip/hip_runtime.h>
#include <stddef.h>

typedef __attribute__((ext_vector_type(2))) float v2f;
typedef __attribute__((ext_vector_type(4))) float v4f;
typedef __attribute__((ext_vector_type(8))) float v8f;

#define IN_F   4096
#define OUT_F  4096
#define NROWS  256      // B*S = 4*64
#define NNZ    262144

// ---------------------------------------------------------------------------
// Kernel 1: zero the dense-W workspace (4096*4096 fp32 = 64 MB)
// ---------------------------------------------------------------------------
__global__ void holo_zero_w(float* __restrict__ W) {
    const size_t n4 = (size_t)IN_F * OUT_F / 4;   // float4 count
    size_t i = (size_t)blockIdx.x * blockDim.x + threadIdx.x;
    const size_t stride = (size_t)gridDim.x * blockDim.x;
    v4f z = {0.f, 0.f, 0.f, 0.f};
    for (; i < n4; i += stride)
        ((v4f*)W)[i] = z;
}

// ---------------------------------------------------------------------------
// Kernel 2: densify — scatter nnz into W[o*IN_F + i] (atomic handles dup coords)
// ---------------------------------------------------------------------------
__global__ void holo_scatter(const int* __restrict__ coords,
                             const float* __restrict__ wts,
                             float* __restrict__ W) {
    int k = blockIdx.x * blockDim.x + threadIdx.x;
    if (k >= NNZ) return;
    int o = coords[2 * k];
    int i = coords[2 * k + 1];
    atomicAdd(W + (size_t)o * IN_F + i, wts[k]);
}

// ---------------------------------------------------------------------------
// Kernel 3: out(256x4096) = X(256x4096) * W^T via V_WMMA_F32_16X16X4_F32.
// One wave -> 16(M) x 64(N) tile: 4 accumulators, shared A fragment.
//   A frag (16x4 f32): lane half h holds K = vgpr + 2h  -> float2 at col 4k+2h
//   B frag (4x16 f32): same K striping; B[k][n] = W[n][k] -> float2 at W[n][4k+2h]
//   D (16x16 f32): vgpr g = row g + 8h, col = lane%16
// ---------------------------------------------------------------------------
__global__ void __launch_bounds__(256)
holo_wmma_gemm(const float* __restrict__ X,
               const float* __restrict__ W,
               float* __restrict__ out) {
    const int wave = threadIdx.x >> 5;
    const int lane = threadIdx.x & 31;
    const int unit = blockIdx.x * 8 + wave;   // 0..1023 wave-tiles
    const int tileM = unit >> 6;              // 16 M-tiles (rows of X)
    const int tileN = unit & 63;              // 64 N-supertiles (cols of out)
    const int m0 = tileM * 16;
    const int n0 = tileN * 64;
    const int lm = lane & 15;
    const int h  = lane >> 4;

    const float* aPtr = X + (size_t)(m0 + lm) * IN_F + 2 * h;
    const float* bPtr = W + (size_t)(n0 + lm) * IN_F + 2 * h;

    v8f c0 = {}, c1 = {}, c2 = {}, c3 = {};

#pragma unroll 4
    for (int kk = 0; kk < IN_F / 4; ++kk) {
        v2f a  = *(const v2f*)(aPtr + kk * 4);
        v2f b0 = *(const v2f*)(bPtr + kk * 4);
        v2f b1 = *(const v2f*)(bPtr + (size_t)16 * IN_F + kk * 4);
        v2f b2 = *(const v2f*)(bPtr + (size_t)32 * IN_F + kk * 4);
        v2f b3 = *(const v2f*)(bPtr + (size_t)48 * IN_F + kk * 4);
        c0 = __builtin_amdgcn_wmma_f32_16x16x4_f32(false, a, false, b0, (short)0, c0, false, false);
        c1 = __builtin_amdgcn_wmma_f32_16x16x4_f32(false, a, false, b1, (short)0, c1, false, false);
        c2 = __builtin_amdgcn_wmma_f32_16x16x4_f32(false, a, false, b2, (short)0, c2, false, false);
        c3 = __builtin_amdgcn_wmma_f32_16x16x4_f32(false, a, false, b3, (short)0, c3, false, false);
    }

    // D layout: vgpr g -> row m0 + g + 8h, col n0 + lm (+16/32/48 per sub-tile)
    float* oBase = out + (size_t)(m0 + 8 * h) * OUT_F + n0 + lm;
#pragma unroll
    for (int g = 0; g < 8; ++g) {
        oBase[(size_t)g * OUT_F +  0] = c0[g];
        oBase[(size_t)g * OUT_F + 16] = c1[g];
        oBase[(size_t)g * OUT_F + 32] = c2[g];
        oBase[(size_t)g * OUT_F + 48] = c3[g];
    }
}

// ---------------------------------------------------------------------------
// Fallback path (only if workspace < 64 MB): direct scatter with atomics.
// ---------------------------------------------------------------------------
__global__ void holo_zero_out(float* __restrict__ out) {
    size_t i = (size_t)blockIdx.x * blockDim.x + threadIdx.x;
    const size_t n = (size_t)NROWS * OUT_F;
    const size_t stride = (size_t)gridDim.x * blockDim.x;
    for (; i < n; i += stride) out[i] = 0.f;
}

__global__ void holo_direct(const float* __restrict__ X,
                            const int* __restrict__ coords,
                            const float* __restrict__ wts,
                            float* __restrict__ out) {
    int k = blockIdx.x * blockDim.x + threadIdx.x;
    if (k >= NNZ) return;
    int o = coords[2 * k];
    int i = coords[2 * k + 1];
    float wv = wts[k];
    for (int r = 0; r < NROWS; ++r)
        atomicAdd(out + (size_t)r * OUT_F + o, X[(size_t)r * IN_F + i] * wv);
}

// ---------------------------------------------------------------------------
extern "C" void kernel_launch(void* const* d_in, const int* in_sizes, int n_in,
                              void* d_out, int out_size, void* d_ws, size_t ws_size,
                              hipStream_t stream) {
    const float* x      = (const float*)d_in[0];
    const int*   coords = (const int*)d_in[1];
    const float* wts    = (const float*)d_in[2];
    float*       out    = (float*)d_out;

    const size_t wBytes = (size_t)IN_F * OUT_F * sizeof(float);  // 64 MB
    if (ws_size >= wBytes) {
        float* W = (float*)d_ws;
        holo_zero_w<<<2048, 256, 0, stream>>>(W);
        holo_scatter<<<NNZ / 256, 256, 0, stream>>>(coords, wts, W);
        holo_wmma_gemm<<<128, 256, 0, stream>>>(x, W, out);   // 1024 waves, full coverage
    } else {
        holo_zero_out<<<1024, 256, 0, stream>>>(out);
        holo_direct<<<NNZ / 256, 256, 0, stream>>>(x, coords, wts, out);
    }
}